// DGLGraphConv_21105469293088
// MI455X (gfx1250) — compile-verified
//
#include <hip/hip_runtime.h>
#include <hip/hip_bf16.h>
#include <math.h>

typedef __attribute__((ext_vector_type(2))) float v2f;
typedef __attribute__((ext_vector_type(8))) float v8f;

#define FEATS 128
#define LDS_STRIDE (FEATS + 4)  // pad: row stride 132 floats (528 B, 16B-aligned)

// ---------------------------------------------------------------------------
// gfx1250 async global->LDS copy (16B per lane), tracked by ASYNCcnt.
// vdst = 32-bit LDS byte offset, vaddr = 64-bit global address, saddr = off.
// ---------------------------------------------------------------------------
static __device__ __forceinline__ void async_copy_b128_to_lds(unsigned lds_off,
                                                              const float* gsrc) {
  asm volatile("global_load_async_to_lds_b128 %0, %1, off"
               :
               : "v"(lds_off), "v"(gsrc)
               : "memory");
}

static __device__ __forceinline__ void wait_async_lds() {
  asm volatile("s_wait_asynccnt 0x0" ::: "memory");
}

// ---------------------------------------------------------------------------
// atomic fp32 multiply via CAS (segment_prod has no native atomic)
// ---------------------------------------------------------------------------
static __device__ __forceinline__ void atomic_mul_f32(float* addr, float val) {
  unsigned int* ua = reinterpret_cast<unsigned int*>(addr);
  unsigned int prev = __float_as_uint(*addr);
  while (true) {
    unsigned int assumed = prev;
    float updated = __uint_as_float(assumed) * val;
    prev = atomicCAS(ua, assumed, __float_as_uint(updated));
    if (prev == assumed) break;
  }
}

// ---------------------------------------------------------------------------
// 1) init: out = 0 (hs accumulator), hp = 1 (prod accumulator), degrees = 0
// ---------------------------------------------------------------------------
__global__ __launch_bounds__(256) void init_kernel(float* __restrict__ out,
                                                   float* __restrict__ hp,
                                                   float* __restrict__ odeg,
                                                   float* __restrict__ ideg,
                                                   int n_nodes) {
  int i = blockIdx.x * 256 + threadIdx.x;
  int total = n_nodes * FEATS;
  if (i < total) {
    out[i] = 0.0f;
    hp[i] = 1.0f;
  }
  if (i < n_nodes) {
    odeg[i] = 0.0f;
    ideg[i] = 0.0f;
  }
}

// ---------------------------------------------------------------------------
// 2) degree accumulation
// ---------------------------------------------------------------------------
__global__ __launch_bounds__(256) void degree_kernel(const int* __restrict__ src,
                                                     const int* __restrict__ dst,
                                                     float* __restrict__ odeg,
                                                     float* __restrict__ ideg,
                                                     int n_edges) {
  int e = blockIdx.x * 256 + threadIdx.x;
  if (e >= n_edges) return;
  atomicAdd(&odeg[src[e]], 1.0f);
  atomicAdd(&ideg[dst[e]], 1.0f);
}

// ---------------------------------------------------------------------------
// 3) degrees -> clip(deg,1)^-1/2 in place
// ---------------------------------------------------------------------------
__global__ __launch_bounds__(256) void deg_finalize_kernel(float* __restrict__ odeg,
                                                           float* __restrict__ ideg,
                                                           int n_nodes) {
  int i = blockIdx.x * 256 + threadIdx.x;
  if (i >= n_nodes) return;
  odeg[i] = rsqrtf(fmaxf(odeg[i], 1.0f));
  ideg[i] = rsqrtf(fmaxf(ideg[i], 1.0f));
}

// ---------------------------------------------------------------------------
// 4) node GEMMs with V_WMMA_F32_16X16X4_F32:
//    h_sum  = feat_src @ w1
//    h_prod = tanh(feat_src @ w2[0:128] + w2[128])   (ones-column bias row)
//    A tile staged via async global->LDS DMA (pure copy); out-degree scale
//    folded into the per-lane A-fragment read (co-executes with XDL WMMA).
//    block = 256 threads = 8 waves; wave w owns output columns [16w,16w+16).
// ---------------------------------------------------------------------------
__global__ __launch_bounds__(256) void node_gemm_kernel(const float* __restrict__ feat,
                                                        const float* __restrict__ w1,
                                                        const float* __restrict__ w2,
                                                        const float* __restrict__ odeg,
                                                        float* __restrict__ h_sum,
                                                        float* __restrict__ h_prod,
                                                        int n_nodes) {
  __shared__ float As[16][LDS_STRIDE];
  const int row0 = blockIdx.x * 16;
  const int tid = threadIdx.x;

  // async-DMA the raw 16x128 feat tile into LDS: 512 x 16B chunks, 2/thread
  const unsigned lds_base = (unsigned)(unsigned long long)(&As[0][0]);
#pragma unroll
  for (int j = 0; j < 2; ++j) {
    int ci = tid + j * 256;      // chunk id 0..511
    int r = ci >> 5;             // 32 chunks per 128-float row
    int c = (ci & 31) * 4;
    int rr = row0 + r;
    if (rr >= n_nodes) rr = n_nodes - 1;  // clamp tail, keep EXEC full
    unsigned loff = lds_base + (unsigned)((r * LDS_STRIDE + c) * 4);
    async_copy_b128_to_lds(loff, feat + (size_t)rr * FEATS + c);
  }
  wait_async_lds();
  __syncthreads();

  const int wave = tid >> 5;
  const int lane = tid & 31;
  const int n0 = wave * 16;
  const int lidx = lane & 15;          // A row in tile / B,D column in tile
  const int khalf = (lane >> 4) * 2;   // lanes 0-15 -> K {k,k+1}; 16-31 -> {k+2,k+3}

  int srow = row0 + lidx;
  if (srow >= n_nodes) srow = n_nodes - 1;
  const float scale = odeg[srow];      // out_deg^-1/2 for this lane's A row

  v8f acc1 = {};
  v8f acc2 = {};
#pragma unroll
  for (int k = 0; k < FEATS; k += 4) {
    v2f a;
    a.x = As[lidx][k + khalf] * scale;
    a.y = As[lidx][k + khalf + 1] * scale;
    const float* w1p = w1 + (size_t)(k + khalf) * FEATS + n0 + lidx;
    const float* w2p = w2 + (size_t)(k + khalf) * FEATS + n0 + lidx;
    v2f b1, b2;
    b1.x = w1p[0];
    b1.y = w1p[FEATS];
    b2.x = w2p[0];
    b2.y = w2p[FEATS];
    acc1 = __builtin_amdgcn_wmma_f32_16x16x4_f32(false, a, false, b1, (short)0, acc1, false, false);
    acc2 = __builtin_amdgcn_wmma_f32_16x16x4_f32(false, a, false, b2, (short)0, acc2, false, false);
  }

  // D layout: VGPR r -> M = r (lanes 0-15) / M = r+8 (lanes 16-31); lane&15 -> N
  const int mhalf = (lane >> 4) * 8;
  const float bias = w2[(size_t)FEATS * FEATS + n0 + lidx];  // w2 row 128
#pragma unroll
  for (int r = 0; r < 8; ++r) {
    int m = row0 + mhalf + r;
    if (m < n_nodes) {
      int n = n0 + lidx;
      h_sum[(size_t)m * FEATS + n] = acc1[r];
      h_prod[(size_t)m * FEATS + n] = tanhf(acc2[r] + bias);
    }
  }
}

// ---------------------------------------------------------------------------
// 5) edge phase: one wave per edge, 4 floats per lane.
//    hs (into d_out) via atomic fp32 add; hp via CAS multiply. All accumulator
//    arrays (51 MB each) are L2-resident on the 192 MB L2.
// ---------------------------------------------------------------------------
__global__ __launch_bounds__(256) void edge_kernel(const int* __restrict__ src,
                                                   const int* __restrict__ dst,
                                                   const int* __restrict__ eattr,
                                                   const float* __restrict__ h_sum,
                                                   const float* __restrict__ h_prod,
                                                   const float* __restrict__ bemb,
                                                   float* __restrict__ hs_out,
                                                   float* __restrict__ hp,
                                                   int n_edges) {
  int wave = (blockIdx.x * 256 + threadIdx.x) >> 5;
  if (wave >= n_edges) return;
  int lane = threadIdx.x & 31;
  int s = src[wave];
  int d = dst[wave];
  int t = eattr[wave];
  int c = lane * 4;

  const float4 ew = *(const float4*)(bemb + (size_t)t * FEATS + c);
  const float4 ps = *(const float4*)(h_sum + (size_t)s * FEATS + c);
  const float4 pp = *(const float4*)(h_prod + (size_t)s * FEATS + c);

  float* so = hs_out + (size_t)d * FEATS + c;
  float* po = hp + (size_t)d * FEATS + c;

  atomicAdd(so + 0, ps.x * ew.x);
  atomicAdd(so + 1, ps.y * ew.y);
  atomicAdd(so + 2, ps.z * ew.z);
  atomicAdd(so + 3, ps.w * ew.w);

  atomic_mul_f32(po + 0, pp.x * ew.x);
  atomic_mul_f32(po + 1, pp.y * ew.y);
  atomic_mul_f32(po + 2, pp.z * ew.z);
  atomic_mul_f32(po + 3, pp.w * ew.w);
}

// ---------------------------------------------------------------------------
// 6) rst = (hs + hp @ v) * in_deg^-1/2, in place on d_out (hs accumulator)
// ---------------------------------------------------------------------------
__global__ __launch_bounds__(256) void final_gemm_kernel(const float* __restrict__ hp,
                                                         const float* __restrict__ vmat,
                                                         const float* __restrict__ ideg,
                                                         float* __restrict__ out,
                                                         int n_nodes) {
  __shared__ float As[16][LDS_STRIDE];
  const int row0 = blockIdx.x * 16;
  const int tid = threadIdx.x;

  // async-DMA the 16x128 hp tile into LDS (pure copy)
  const unsigned lds_base = (unsigned)(unsigned long long)(&As[0][0]);
#pragma unroll
  for (int j = 0; j < 2; ++j) {
    int ci = tid + j * 256;
    int r = ci >> 5;
    int c = (ci & 31) * 4;
    int rr = row0 + r;
    if (rr >= n_nodes) rr = n_nodes - 1;
    unsigned loff = lds_base + (unsigned)((r * LDS_STRIDE + c) * 4);
    async_copy_b128_to_lds(loff, hp + (size_t)rr * FEATS + c);
  }
  wait_async_lds();
  __syncthreads();

  const int wave = tid >> 5;
  const int lane = tid & 31;
  const int n0 = wave * 16;
  const int lidx = lane & 15;
  const int khalf = (lane >> 4) * 2;

  v8f acc = {};
#pragma unroll
  for (int k = 0; k < FEATS; k += 4) {
    v2f a;
    a.x = As[lidx][k + khalf];
    a.y = As[lidx][k + khalf + 1];
    const float* vp = vmat + (size_t)(k + khalf) * FEATS + n0 + lidx;
    v2f b;
    b.x = vp[0];
    b.y = vp[FEATS];
    acc = __builtin_amdgcn_wmma_f32_16x16x4_f32(false, a, false, b, (short)0, acc, false, false);
  }

  const int mhalf = (lane >> 4) * 8;
#pragma unroll
  for (int r = 0; r < 8; ++r) {
    int m = row0 + mhalf + r;
    if (m < n_nodes) {
      int n = n0 + lidx;
      size_t idx = (size_t)m * FEATS + n;
      out[idx] = (out[idx] + acc[r]) * ideg[m];
    }
  }
}

// ---------------------------------------------------------------------------
// launch
// ---------------------------------------------------------------------------
extern "C" void kernel_launch(void* const* d_in, const int* in_sizes, int n_in,
                              void* d_out, int out_size, void* d_ws, size_t ws_size,
                              hipStream_t stream) {
  const float* feat = (const float*)d_in[0];
  const int* src = (const int*)d_in[1];
  const int* dst = (const int*)d_in[2];
  const int* eattr = (const int*)d_in[3];
  const float* w1 = (const float*)d_in[4];
  const float* w2 = (const float*)d_in[5];  // (129, 128): 128 matrix rows + bias row
  const float* v = (const float*)d_in[6];
  const float* bemb = (const float*)d_in[7];
  float* out = (float*)d_out;

  const int N = in_sizes[0] / FEATS;
  const int E = in_sizes[1];

  float* h_sum = (float*)d_ws;
  float* h_prod = h_sum + (size_t)N * FEATS;
  float* hp = h_prod + (size_t)N * FEATS;
  float* odeg = hp + (size_t)N * FEATS;
  float* ideg = odeg + N;

  const int total = N * FEATS;
  init_kernel<<<(total + 255) / 256, 256, 0, stream>>>(out, hp, odeg, ideg, N);
  degree_kernel<<<(E + 255) / 256, 256, 0, stream>>>(src, dst, odeg, ideg, E);
  deg_finalize_kernel<<<(N + 255) / 256, 256, 0, stream>>>(odeg, ideg, N);
  node_gemm_kernel<<<(N + 15) / 16, 256, 0, stream>>>(feat, w1, w2, odeg, h_sum, h_prod, N);
  edge_kernel<<<(int)(((size_t)E * 32 + 255) / 256), 256, 0, stream>>>(
      src, dst, eattr, h_sum, h_prod, bemb, out, hp, E);
  final_gemm_kernel<<<(N + 15) / 16, 256, 0, stream>>>(hp, v, ideg, out, N);
}